// GAT_27041114096323
// MI455X (gfx1250) — compile-verified
//
#include <hip/hip_runtime.h>
#include <hip/hip_bf16.h>

// ---------------------------------------------------------------------------
// 2-layer GAT for MI455X (gfx1250, wave32).
// GEMMs: V_WMMA_F32_16X16X4_F32 (fp32-exact, matches fp32 reference),
//        2x 16-row blocks per wave (B-fragment reuse), b64 A loads,
//        __launch_bounds__(128,1) so 128+ accumulator VGPRs stay resident
//        (round-2 build spilled accumulators to scratch in the k-loop).
// Edge phase (roofline-limiting ~2.5GB irregular traffic @ 23.3TB/s HBM):
//        float4 (b128) gathers + native f32 global atomics.
// ---------------------------------------------------------------------------

typedef __attribute__((ext_vector_type(2))) float v2f;
typedef __attribute__((ext_vector_type(8))) float v8f;

#define NEG_SLOPE 0.2f
#define EPSF 1e-16f

__device__ __forceinline__ v8f wmma_f32(v2f a, v2f b, v8f c) {
  // (neg_a, A, neg_b, B, c_mod, C, reuse_a, reuse_b)
  return __builtin_amdgcn_wmma_f32_16x16x4_f32(
      false, a, false, b, (short)0, c, false, false);
}

// --- generic 32-bit fill (re-init accumulators every call) -----------------
__global__ void fill_u32_k(unsigned int* __restrict__ p, unsigned int v, long long n) {
  long long i = (long long)blockIdx.x * blockDim.x + threadIdx.x;
  if (i < n) p[i] = v;
}

// --- GEMM1: H1[N,128] = X[N,128] @ W1[128,128] -----------------------------
// One wave handles two 16-row blocks (32 rows): each B fragment is loaded
// once per k-step and fed to two WMMAs. A fragments load as b64 (v2f).
__global__ void __launch_bounds__(128, 1)
gemm1_wmma_k(const float* __restrict__ X, const float* __restrict__ W,
             float* __restrict__ Hout, int N) {
  int wave = blockIdx.x * (blockDim.x >> 5) + (threadIdx.x >> 5);
  int lane = threadIdx.x & 31;
  int half = lane >> 4;
  int lm   = lane & 15;
  long long row0 = (long long)wave * 32;
  if (row0 >= N) return;                    // wave-uniform: EXEC all-ones at WMMA
  long long r0 = row0 + lm;
  long long r1 = row0 + 16 + lm;
  bool ok0 = r0 < N, ok1 = r1 < N;
  const v2f* x0 = (const v2f*)(X + r0 * 128);
  const v2f* x1 = (const v2f*)(X + r1 * 128);
  const v2f zz = {0.0f, 0.0f};

  v8f acc0[8] = {}, acc1[8] = {};
  for (int k = 0; k < 128; k += 4) {
    int ka = k + 2 * half;                  // A 16x4 f32: VGPR0={K0|K2}, VGPR1={K1|K3}
    v2f a0 = ok0 ? x0[ka >> 1] : zz;        // one global_load_b64 each
    v2f a1 = ok1 ? x1[ka >> 1] : zz;
    const float* wr = W + (long long)ka * 128;
#pragma unroll
    for (int t = 0; t < 8; ++t) {
      int col = t * 16 + lm;
      v2f b = {wr[col], wr[128 + col]};     // B 4x16, K-pair split across halves
      acc0[t] = wmma_f32(a0, b, acc0[t]);
      acc1[t] = wmma_f32(a1, b, acc1[t]);
    }
  }
#pragma unroll
  for (int t = 0; t < 8; ++t) {
    int col = t * 16 + lm;
#pragma unroll
    for (int i = 0; i < 8; ++i) {           // C/D: VGPR i -> M = i (+8 for hi half)
      long long ra = row0 + i + 8 * half;
      long long rb = ra + 16;
      if (ra < N) Hout[ra * 128 + col] = acc0[t][i];
      if (rb < N) Hout[rb * 128 + col] = acc1[t][i];
    }
  }
}

// --- GEMM2: G[N,40] = H2[N,128] @ W2[128,40] (3 col tiles, guarded) --------
__global__ void __launch_bounds__(128, 1)
gemm2_wmma_k(const float* __restrict__ X, const float* __restrict__ W,
             float* __restrict__ G, int N) {
  int wave = blockIdx.x * (blockDim.x >> 5) + (threadIdx.x >> 5);
  int lane = threadIdx.x & 31;
  int half = lane >> 4;
  int lm   = lane & 15;
  long long row0 = (long long)wave * 32;
  if (row0 >= N) return;                    // wave-uniform
  long long r0 = row0 + lm;
  long long r1 = row0 + 16 + lm;
  bool ok0 = r0 < N, ok1 = r1 < N;
  const v2f* x0 = (const v2f*)(X + r0 * 128);
  const v2f* x1 = (const v2f*)(X + r1 * 128);
  const v2f zz = {0.0f, 0.0f};

  v8f acc0[3] = {}, acc1[3] = {};
  for (int k = 0; k < 128; k += 4) {
    int ka = k + 2 * half;
    v2f a0 = ok0 ? x0[ka >> 1] : zz;
    v2f a1 = ok1 ? x1[ka >> 1] : zz;
    const float* wr = W + (long long)ka * 40;
#pragma unroll
    for (int t = 0; t < 3; ++t) {
      int col = t * 16 + lm;
      v2f b;
      b.x = (col < 40) ? wr[col]      : 0.0f;
      b.y = (col < 40) ? wr[40 + col] : 0.0f;
      acc0[t] = wmma_f32(a0, b, acc0[t]);
      acc1[t] = wmma_f32(a1, b, acc1[t]);
    }
  }
#pragma unroll
  for (int t = 0; t < 3; ++t) {
    int col = t * 16 + lm;
    if (col < 40) {
#pragma unroll
      for (int i = 0; i < 8; ++i) {
        long long ra = row0 + i + 8 * half;
        long long rb = ra + 16;
        if (ra < N) G[ra * 40 + col] = acc0[t][i];
        if (rb < N) G[rb * 40 + col] = acc1[t][i];
      }
    }
  }
}

// --- attention coefficients layer 1: [N,8] each (float4 reads) -------------
__global__ void alpha1_k(const float* __restrict__ H1, const float* __restrict__ asrc,
                         const float* __restrict__ adst, float* __restrict__ AS,
                         float* __restrict__ AD, int N) {
  long long tid = (long long)blockIdx.x * blockDim.x + threadIdx.x;
  if (tid >= (long long)N * 8) return;
  int h = (int)(tid & 7);
  const float4* hv = (const float4*)(H1 + (tid >> 3) * 128 + h * 16);  // 64B aligned
  const float4* av = (const float4*)(asrc + h * 16);
  const float4* dv = (const float4*)(adst + h * 16);
  float s = 0.f, d = 0.f;
#pragma unroll
  for (int q = 0; q < 4; ++q) {
    float4 v = hv[q], a = av[q], b = dv[q];
    s += v.x * a.x + v.y * a.y + v.z * a.z + v.w * a.w;
    d += v.x * b.x + v.y * b.y + v.z * b.z + v.w * b.w;
  }
  AS[tid] = s; AD[tid] = d;
}

// --- attention coefficients layer 2: [N] each ------------------------------
__global__ void alpha2_k(const float* __restrict__ G, const float* __restrict__ asrc,
                         const float* __restrict__ adst, float* __restrict__ AS,
                         float* __restrict__ AD, int N) {
  long long n = (long long)blockIdx.x * blockDim.x + threadIdx.x;
  if (n >= N) return;
  const float4* gv = (const float4*)(G + n * 40);   // n*160B -> 16B aligned
  float s = 0.f, d = 0.f;
#pragma unroll
  for (int q = 0; q < 10; ++q) {
    float4 v = gv[q];
    float4 a = ((const float4*)asrc)[q];
    float4 b = ((const float4*)adst)[q];
    s += v.x * a.x + v.y * a.y + v.z * a.z + v.w * a.w;
    d += v.x * b.x + v.y * b.y + v.z * b.z + v.w * b.w;
  }
  AS[n] = s; AD[n] = d;
}

__device__ __forceinline__ float lrelu(float v) { return v > 0.f ? v : NEG_SLOPE * v; }

// sign-split atomic float max (init = -inf bits 0xFF800000)
__device__ __forceinline__ void atomicMaxF(float* addr, float v) {
  if (v >= 0.f) atomicMax((int*)addr, __float_as_int(v));
  else          atomicMin((unsigned int*)addr, __float_as_uint(v));
}

__device__ __forceinline__ void edge_sd(long long e, long long E,
                                        const int* __restrict__ src,
                                        const int* __restrict__ dst,
                                        int& s, int& d) {
  if (e < E) { s = src[e]; d = dst[e]; }
  else       { s = (int)(e - E); d = s; }   // appended self-loops
}

// --- layer1 softmax: max / exp+sum / message scatter ------------------------
__global__ void edge_max1_k(const int* __restrict__ src, const int* __restrict__ dst,
                            const float* __restrict__ AS, const float* __restrict__ AD,
                            float* __restrict__ EMAX, long long E, long long E2) {
  long long tid = (long long)blockIdx.x * blockDim.x + threadIdx.x;
  if (tid >= E2 * 8) return;
  long long e = tid >> 3; int h = (int)(tid & 7);
  int s, d; edge_sd(e, E, src, dst, s, d);
  float v = lrelu(AS[(long long)s * 8 + h] + AD[(long long)d * 8 + h]);
  atomicMaxF(&EMAX[(long long)d * 8 + h], v);
}

__global__ void edge_exp1_k(const int* __restrict__ src, const int* __restrict__ dst,
                            const float* __restrict__ AS, const float* __restrict__ AD,
                            const float* __restrict__ EMAX, float* __restrict__ EBUF,
                            float* __restrict__ DEN, long long E, long long E2) {
  long long tid = (long long)blockIdx.x * blockDim.x + threadIdx.x;
  if (tid >= E2 * 8) return;
  long long e = tid >> 3; int h = (int)(tid & 7);
  int s, d; edge_sd(e, E, src, dst, s, d);
  float v = lrelu(AS[(long long)s * 8 + h] + AD[(long long)d * 8 + h]);
  float ex = __expf(v - EMAX[(long long)d * 8 + h]);
  EBUF[tid] = ex;
  atomicAdd(&DEN[(long long)d * 8 + h], ex);
}

__global__ void edge_msg1_k(const int* __restrict__ src, const int* __restrict__ dst,
                            const float* __restrict__ H1, const float* __restrict__ EBUF,
                            const float* __restrict__ DEN, float* __restrict__ ACC,
                            long long E, long long E2) {
  long long tid = (long long)blockIdx.x * blockDim.x + threadIdx.x;
  if (tid >= E2 * 8) return;
  long long e = tid >> 3; int h = (int)(tid & 7);
  int s, d; edge_sd(e, E, src, dst, s, d);
  float alpha = EBUF[tid] / (DEN[(long long)d * 8 + h] + EPSF);
  const float4* hv = (const float4*)(H1 + (long long)s * 128 + h * 16);  // 4x b128
  float4 v0 = hv[0], v1 = hv[1], v2 = hv[2], v3 = hv[3];
  float* ac = ACC + (long long)d * 128 + h * 16;
  atomicAdd(&ac[0],  v0.x * alpha); atomicAdd(&ac[1],  v0.y * alpha);
  atomicAdd(&ac[2],  v0.z * alpha); atomicAdd(&ac[3],  v0.w * alpha);
  atomicAdd(&ac[4],  v1.x * alpha); atomicAdd(&ac[5],  v1.y * alpha);
  atomicAdd(&ac[6],  v1.z * alpha); atomicAdd(&ac[7],  v1.w * alpha);
  atomicAdd(&ac[8],  v2.x * alpha); atomicAdd(&ac[9],  v2.y * alpha);
  atomicAdd(&ac[10], v2.z * alpha); atomicAdd(&ac[11], v2.w * alpha);
  atomicAdd(&ac[12], v3.x * alpha); atomicAdd(&ac[13], v3.y * alpha);
  atomicAdd(&ac[14], v3.z * alpha); atomicAdd(&ac[15], v3.w * alpha);
}

// --- ELU(acc + b1), in place, float4 ---------------------------------------
__global__ void elu_bias1_k(float* __restrict__ ACC, const float* __restrict__ b1, long long n4) {
  long long tid = (long long)blockIdx.x * blockDim.x + threadIdx.x;
  if (tid >= n4) return;
  float4 v = ((const float4*)ACC)[tid];
  float4 b = ((const float4*)b1)[tid & 31];   // (tid*4) % 128 / 4
  v.x += b.x; v.y += b.y; v.z += b.z; v.w += b.w;
  v.x = v.x > 0.f ? v.x : (__expf(v.x) - 1.f);
  v.y = v.y > 0.f ? v.y : (__expf(v.y) - 1.f);
  v.z = v.z > 0.f ? v.z : (__expf(v.z) - 1.f);
  v.w = v.w > 0.f ? v.w : (__expf(v.w) - 1.f);
  ((float4*)ACC)[tid] = v;
}

// --- layer2 softmax + scatter ----------------------------------------------
__global__ void edge_max2_k(const int* __restrict__ src, const int* __restrict__ dst,
                            const float* __restrict__ AS, const float* __restrict__ AD,
                            float* __restrict__ EMAX, long long E, long long E2) {
  long long e = (long long)blockIdx.x * blockDim.x + threadIdx.x;
  if (e >= E2) return;
  int s, d; edge_sd(e, E, src, dst, s, d);
  atomicMaxF(&EMAX[d], lrelu(AS[s] + AD[d]));
}

__global__ void edge_exp2_k(const int* __restrict__ src, const int* __restrict__ dst,
                            const float* __restrict__ AS, const float* __restrict__ AD,
                            const float* __restrict__ EMAX, float* __restrict__ EBUF,
                            float* __restrict__ DEN, long long E, long long E2) {
  long long e = (long long)blockIdx.x * blockDim.x + threadIdx.x;
  if (e >= E2) return;
  int s, d; edge_sd(e, E, src, dst, s, d);
  float ex = __expf(lrelu(AS[s] + AD[d]) - EMAX[d]);
  EBUF[e] = ex;
  atomicAdd(&DEN[d], ex);
}

__global__ void edge_msg2_k(const int* __restrict__ src, const int* __restrict__ dst,
                            const float* __restrict__ G, const float* __restrict__ EBUF,
                            const float* __restrict__ DEN, float* __restrict__ OUT,
                            long long E, long long E2) {
  long long tid = (long long)blockIdx.x * blockDim.x + threadIdx.x;
  if (tid >= E2 * 40) return;
  long long e = tid / 40; int c = (int)(tid - e * 40);
  int s, d; edge_sd(e, E, src, dst, s, d);
  float alpha = EBUF[e] / (DEN[d] + EPSF);
  atomicAdd(&OUT[(long long)d * 40 + c], G[(long long)s * 40 + c] * alpha);
}

__global__ void bias2_k(float* __restrict__ OUT, const float* __restrict__ b2, long long n4) {
  long long tid = (long long)blockIdx.x * blockDim.x + threadIdx.x;
  if (tid >= n4) return;
  float4 v = ((const float4*)OUT)[tid];
  float4 b = ((const float4*)b2)[tid % 10];   // (tid*4) % 40 / 4
  v.x += b.x; v.y += b.y; v.z += b.z; v.w += b.w;
  ((float4*)OUT)[tid] = v;
}

// ---------------------------------------------------------------------------
static inline unsigned int nblk(long long n, int b) { return (unsigned int)((n + b - 1) / b); }

extern "C" void kernel_launch(void* const* d_in, const int* in_sizes, int n_in,
                              void* d_out, int out_size, void* d_ws, size_t ws_size,
                              hipStream_t stream) {
  const float* x      = (const float*)d_in[0];
  const int*   ei     = (const int*)d_in[1];   // [2,E] int32 (JAX demotes int64)
  const float* W1     = (const float*)d_in[2];
  const float* a_src1 = (const float*)d_in[3];
  const float* a_dst1 = (const float*)d_in[4];
  const float* b1     = (const float*)d_in[5];
  const float* W2     = (const float*)d_in[6];
  const float* a_src2 = (const float*)d_in[7];
  const float* a_dst2 = (const float*)d_in[8];
  const float* b2     = (const float*)d_in[9];
  float*       out    = (float*)d_out;

  const long long N  = in_sizes[0] / 128;
  const long long E  = in_sizes[1] / 2;
  const long long E2 = E + N;
  const int* srcA = ei;
  const int* dstA = ei + E;

  // --- workspace carve-up (256B aligned blocks) ---
  char* ws = (char*)d_ws;
  size_t off = 0;
  auto carve = [&](size_t bytes) -> void* {
    void* p = ws + off;
    off = (off + bytes + 255) & ~(size_t)255;
    return p;
  };
  float* H1    = (float*)carve((size_t)N * 128 * 4);   // x @ W1
  float* AS1   = (float*)carve((size_t)N * 8 * 4);
  float* AD1   = (float*)carve((size_t)N * 8 * 4);
  float* EMAX1 = (float*)carve((size_t)N * 8 * 4);
  float* DEN1  = (float*)carve((size_t)N * 8 * 4);
  float* EB1   = (float*)carve((size_t)E2 * 8 * 4);    // per-edge exp, layer 1
  float* ACC1  = (float*)carve((size_t)N * 128 * 4);   // layer-1 agg -> ELU -> h2
  float* G2    = (float*)carve((size_t)N * 40 * 4);    // h2 @ W2
  float* AS2   = (float*)carve((size_t)N * 4);
  float* AD2   = (float*)carve((size_t)N * 4);
  float* EMAX2 = (float*)carve((size_t)N * 4);
  float* DEN2  = (float*)carve((size_t)N * 4);
  float* EB2   = (float*)carve((size_t)E2 * 4);        // per-edge exp, layer 2

  const unsigned int NEG_INF = 0xFF800000u;
  const int B = 256;

  // --- init accumulators (ws/d_out poisoned; re-init every call) ---
  fill_u32_k<<<nblk(N * 8, B), B, 0, stream>>>((unsigned int*)EMAX1, NEG_INF, N * 8);
  fill_u32_k<<<nblk(N * 8, B), B, 0, stream>>>((unsigned int*)DEN1, 0u, N * 8);
  fill_u32_k<<<nblk(N * 128, B), B, 0, stream>>>((unsigned int*)ACC1, 0u, N * 128);
  fill_u32_k<<<nblk(N, B), B, 0, stream>>>((unsigned int*)EMAX2, NEG_INF, N);
  fill_u32_k<<<nblk(N, B), B, 0, stream>>>((unsigned int*)DEN2, 0u, N);
  fill_u32_k<<<nblk(N * 40, B), B, 0, stream>>>((unsigned int*)out, 0u, N * 40);

  // --- layer 1 ---
  {
    long long waves = (N + 31) / 32;                 // one wave per 32-row block
    gemm1_wmma_k<<<nblk(waves, 4), 128, 0, stream>>>(x, W1, H1, (int)N);
  }
  alpha1_k<<<nblk(N * 8, B), B, 0, stream>>>(H1, a_src1, a_dst1, AS1, AD1, (int)N);
  edge_max1_k<<<nblk(E2 * 8, B), B, 0, stream>>>(srcA, dstA, AS1, AD1, EMAX1, E, E2);
  edge_exp1_k<<<nblk(E2 * 8, B), B, 0, stream>>>(srcA, dstA, AS1, AD1, EMAX1, EB1, DEN1, E, E2);
  edge_msg1_k<<<nblk(E2 * 8, B), B, 0, stream>>>(srcA, dstA, H1, EB1, DEN1, ACC1, E, E2);
  elu_bias1_k<<<nblk(N * 32, B), B, 0, stream>>>(ACC1, b1, N * 32);   // N*128/4

  // --- layer 2 ---
  {
    long long waves = (N + 31) / 32;
    gemm2_wmma_k<<<nblk(waves, 4), 128, 0, stream>>>(ACC1, W2, G2, (int)N);
  }
  alpha2_k<<<nblk(N, B), B, 0, stream>>>(G2, a_src2, a_dst2, AS2, AD2, (int)N);
  edge_max2_k<<<nblk(E2, B), B, 0, stream>>>(srcA, dstA, AS2, AD2, EMAX2, E, E2);
  edge_exp2_k<<<nblk(E2, B), B, 0, stream>>>(srcA, dstA, AS2, AD2, EMAX2, EB2, DEN2, E, E2);
  edge_msg2_k<<<nblk(E2 * 40, B), B, 0, stream>>>(srcA, dstA, G2, EB2, DEN2, out, E, E2);
  bias2_k<<<nblk(N * 10, B), B, 0, stream>>>(out, b2, N * 10);        // N*40/4
}